// FeatureExpander_79130477461507
// MI455X (gfx1250) — compile-verified
//
#include <hip/hip_runtime.h>
#include <cstdint>

// ---------------------------------------------------------------------------
// FeatureExpander for MI455X (gfx1250, wave32).
//
// Pure bandwidth problem: 52 MB in, 891 MB out -> ~40us floor at 23.3 TB/s.
// WMMA is inapplicable (x_i*x_j per row has no K-contraction), so we optimize
// data movement:
//   - input tile staged via CDNA5 async copy: global_load_async_to_lds_b128
//     + s_wait_asynccnt (ASYNCcnt path)
//   - output written as b128 non-temporal streaming stores (891 MB >> 192 MB
//     L2, so NT avoids rinsing the L2)
//   - per-column work in a 425-entry LDS descriptor table; copy/x^2/pair ops
//     all collapse onto ONE straight-line path: v = a * (op ? b : 1.0f),
//     covering 350/425 columns with 2 LDS reads + cndmask + v_mul_f32.
// ---------------------------------------------------------------------------

#define NCOL    25
#define NFEAT   425            // 5*25 + 300
#define ROWS    16             // rows per block; tile = 16*25 floats = 1600 B
#define THREADS 256            // 8 wave32 waves

// Descriptor per output column: (op<<16) | (i<<8) | j
//   op 0 = copy x[i]            (j = 0, unused -> b read is harmless)
//   op 1 = x[i] * x[j]          (covers x^2 when i==j, and interactions)
//   op 2 = log(|x[i]|+eps)
//   op 3 = sqrt(|x[i]|+eps)
//   op 4 = (|x[i]|+eps)^(1/3)
struct DescTab { unsigned d[NFEAT]; };
static constexpr DescTab makeDesc() {
    DescTab t{};
    for (int c = 0; c < NCOL; ++c) t.d[c]       = (0u << 16) | (unsigned)(c << 8);
    for (int c = 0; c < NCOL; ++c) t.d[25 + c]  = (1u << 16) | (unsigned)(c << 8) | (unsigned)c;
    for (int c = 0; c < NCOL; ++c) t.d[50 + c]  = (2u << 16) | (unsigned)(c << 8);
    for (int c = 0; c < NCOL; ++c) t.d[75 + c]  = (3u << 16) | (unsigned)(c << 8);
    for (int c = 0; c < NCOL; ++c) t.d[100 + c] = (4u << 16) | (unsigned)(c << 8);
    int k = 125;  // np.triu_indices(25, k=1) row-major order
    for (int i = 0; i < NCOL; ++i)
        for (int j = i + 1; j < NCOL; ++j)
            t.d[k++] = (1u << 16) | (unsigned)(i << 8) | (unsigned)j;
    return t;
}
__constant__ DescTab cDesc = makeDesc();

typedef __attribute__((ext_vector_type(4))) float v4f;

__device__ __forceinline__ float featval(const float* __restrict__ sx,
                                         const unsigned* __restrict__ sdesc,
                                         int row, int cc) {
    const unsigned d  = sdesc[cc];                 // LDS read (lane-consecutive)
    const float*   xr = sx + row * NCOL;
    const float    a  = xr[(d >> 8) & 0xFF];       // LDS broadcast read
    const float    b  = xr[d & 0xFF];              // LDS broadcast read
    const unsigned op = d >> 16;
    if (op <= 1) {
        // copy and mul share one path: 350/425 columns, branch-free body.
        return a * (op ? b : 1.0f);
    }
    const float ax = __builtin_fabsf(a) + 1e-10f;
    const float lg = __logf(ax);
    // One divergence region for the 75 transcendental columns.
    return (op == 2) ? lg
         : (op == 3) ? __builtin_sqrtf(ax)
                     : __expf(lg * (1.0f / 3.0f));  // jnp.power(ax,1/3)
}

__global__ __launch_bounds__(THREADS)
void feature_expander_kernel(const float* __restrict__ x,
                             float* __restrict__ out,
                             int B) {
    __shared__ float    sx[ROWS * NCOL];   // 1600 B
    __shared__ unsigned sdesc[NFEAT];      // 1700 B

    const int tid      = threadIdx.x;
    const int rowBase  = blockIdx.x * ROWS;
    const int rowsHere = min(ROWS, B - rowBase);
    const float* gsrc  = x + (size_t)rowBase * NCOL;

    // Stage descriptor table into LDS (2 entries per thread).
    for (int i = tid; i < NFEAT; i += THREADS) sdesc[i] = cDesc.d[i];

    if (rowsHere == ROWS) {
        // Full tile: rows contiguous -> the 16x25 tile is one contiguous,
        // 16 B-aligned 1600 B span. Stage via 100 async b128 global->LDS DMAs.
        if (tid < (ROWS * NCOL) / 4) {
            unsigned      ldsAddr = (unsigned)(uintptr_t)&sx[tid * 4];
            unsigned long gaddr   = (unsigned long)(gsrc + tid * 4);
            asm volatile("global_load_async_to_lds_b128 %0, %1, off"
                         :: "v"(ldsAddr), "v"(gaddr)
                         : "memory");
        }
        asm volatile("s_wait_asynccnt 0" ::: "memory");
    } else {
        // Tail block (doesn't occur for B=524288; kept for generality).
        for (int i = tid; i < rowsHere * NCOL; i += THREADS) sx[i] = gsrc[i];
    }
    __syncthreads();

    float* gout = out + (size_t)rowBase * NFEAT;   // contiguous output tile

    if (rowsHere == ROWS) {
        // 6800 outputs = 1700 float4 chunks; 16 B-aligned b128 NT stores.
        const int NCHUNK = (ROWS * NFEAT) / 4;     // 1700
        for (int c = tid; c < NCHUNK; c += THREADS) {
            const int gidx = c * 4;
            const int row  = (int)((unsigned)gidx / NFEAT);  // magic-mul div
            const int col  = gidx - row * NFEAT;
            v4f q;
#pragma unroll
            for (int e = 0; e < 4; ++e) {
                int rr = row, cc = col + e;
                if (cc >= NFEAT) { cc -= NFEAT; ++rr; }      // at most one wrap
                q[e] = featval(sx, sdesc, rr, cc);
            }
            __builtin_nontemporal_store(q, (v4f*)(gout + gidx));
        }
    } else {
        // Scalar path for a partial tile.
        const int nOut = rowsHere * NFEAT;
        for (int gidx = tid; gidx < nOut; gidx += THREADS) {
            const int row = (int)((unsigned)gidx / NFEAT);
            const int cc  = gidx - row * NFEAT;
            __builtin_nontemporal_store(featval(sx, sdesc, row, cc), gout + gidx);
        }
    }
}

extern "C" void kernel_launch(void* const* d_in, const int* in_sizes, int n_in,
                              void* d_out, int out_size, void* d_ws, size_t ws_size,
                              hipStream_t stream) {
    const float* x = (const float*)d_in[0];
    float* out     = (float*)d_out;

    const int B      = in_sizes[0] / NCOL;        // 524288
    const int blocks = (B + ROWS - 1) / ROWS;     // 32768

    feature_expander_kernel<<<dim3(blocks), dim3(THREADS), 0, stream>>>(x, out, B);
}